// MoT_43533788512463
// MI455X (gfx1250) — compile-verified
//
#include <hip/hip_runtime.h>

// CDNA5 / gfx1250, wave32.
typedef __attribute__((ext_vector_type(2))) float v2f;
typedef __attribute__((ext_vector_type(8))) float v8f;

#define MDIM 8      // heads
#define KDIM 32     // embedding dim per head
#define MK   256    // M*K floats per user row

// One edge per wave32.
//   A-matrix (16x32 f32): rows 0-7 = attn_emb[u], rows 8-15 = taste_emb[u]
//   B-matrix (32x16 f32): movie_emb[v] replicated into all 16 columns
//   D = A x B (chained 8x V_WMMA_F32_16X16X4_F32, f32 accumulate)
//   -> lanes 0-15 hold logits[0..7] in acc[0..7], lanes 16-31 hold scores[0..7]
__global__ __launch_bounds__(256) void mot_wmma_kernel(
    const int*   __restrict__ edge,     // (B,2) int32
    const float* __restrict__ taste,    // (N_USERS, 256)
    const float* __restrict__ attn,     // (N_USERS, 256)
    const float* __restrict__ movie,    // (N_MOVIES, 32)
    const float* __restrict__ ubias,    // (N_USERS, 1)
    const float* __restrict__ mbias,    // (N_MOVIES, 1)
    float*       __restrict__ out,      // (B,)
    int nB)
{
    const int lane = threadIdx.x & 31;
    const int wave = blockIdx.x * (blockDim.x >> 5) + (threadIdx.x >> 5);
    if (wave >= nB) return;            // wave-uniform: EXEC stays all-1s

    const int u = edge[2 * wave + 0];
    const int v = edge[2 * wave + 1];

    const int half = lane >> 4;        // 0: K sub-offsets {0,1}; 1: {2,3}
    const int row  = lane & 15;        // A-matrix row this lane feeds
    const int kb   = half * 2;

    // Row pointer into the stacked [attn; taste] 16x32 A-matrix.
    const float* aurow = (row < 8)
        ? (attn  + (size_t)u * MK + (size_t)row       * KDIM)
        : (taste + (size_t)u * MK + (size_t)(row - 8) * KDIM);
    const float* erow = movie + (size_t)v * KDIM;

    v8f acc = {};
#pragma unroll
    for (int j = 0; j < 8; ++j) {
        // A frag: vgpr0 = AU[row, 4j+kb], vgpr1 = AU[row, 4j+kb+1]
        v2f a = *(const v2f*)(aurow + 4 * j + kb);
        // B frag: e[k] broadcast across all 16 columns (constant per half-wave)
        v2f b = *(const v2f*)(erow + 4 * j + kb);
        acc = __builtin_amdgcn_wmma_f32_16x16x4_f32(
            /*neg_a=*/false, a, /*neg_b=*/false, b,
            /*c_mod=*/(short)0, acc, /*reuse_a=*/false, /*reuse_b=*/false);
    }

    // Exchange halves so every lane holds all 8 logits and all 8 scores.
    const bool low = (half == 0);
    float logit[MDIM], score[MDIM];
#pragma unroll
    for (int r = 0; r < MDIM; ++r) {
        float mine  = acc[r];
        float other = __shfl_xor(mine, 16, 32);
        logit[r] = low ? mine  : other;
        score[r] = low ? other : mine;
    }

    // Softmax over M=8 + weighted sum (identical in every lane).
    float mx = logit[0];
#pragma unroll
    for (int r = 1; r < MDIM; ++r) mx = fmaxf(mx, logit[r]);
    float Z = 0.f, num = 0.f;
#pragma unroll
    for (int r = 0; r < MDIM; ++r) {
        float w = __expf(logit[r] - mx);
        Z   += w;
        num  = fmaf(w, score[r], num);
    }
    float y = num / Z + mbias[v] + ubias[u];

    if (lane == 0) out[wave] = y;
}

extern "C" void kernel_launch(void* const* d_in, const int* in_sizes, int n_in,
                              void* d_out, int out_size, void* d_ws, size_t ws_size,
                              hipStream_t stream) {
    const int*   edge  = (const int*)  d_in[0];
    const float* taste = (const float*)d_in[1];
    const float* attn  = (const float*)d_in[2];
    const float* movie = (const float*)d_in[3];
    const float* ubias = (const float*)d_in[4];
    const float* mbias = (const float*)d_in[5];
    float*       out   = (float*)d_out;

    const int nB = in_sizes[0] / 2;          // 524288 edges
    const int wavesPerBlock = 256 / 32;      // 8 edges per 256-thread block
    const int blocks = (nB + wavesPerBlock - 1) / wavesPerBlock;
    mot_wmma_kernel<<<blocks, 256, 0, stream>>>(edge, taste, attn, movie,
                                                ubias, mbias, out, nB);
}